// SAGE_encoder_22179211117092
// MI455X (gfx1250) — compile-verified
//
#include <hip/hip_runtime.h>

#define N_NODES 50000
#define D 128
#define BN_EPS 1e-5f
#define L2_EPS 1e-12f

typedef float v2f __attribute__((ext_vector_type(2)));
typedef float v8f __attribute__((ext_vector_type(8)));

// ---------------------------------------------------------------- zero
__global__ void zero_kernel(float* __restrict__ p, size_t n) {
    size_t i = (size_t)blockIdx.x * blockDim.x + threadIdx.x;
    size_t stride = (size_t)gridDim.x * blockDim.x;
    for (; i < n; i += stride) p[i] = 0.0f;
}

// ------------------------------------------------- edge scatter (mean agg)
// one wave per edge: 32 lanes x float4 = 128 floats, coalesced 512B row read
__global__ __launch_bounds__(256)
void scatter_kernel(const float* __restrict__ feat,
                    const long long* __restrict__ src,
                    const long long* __restrict__ dst,
                    float* __restrict__ agg, float* __restrict__ cnt, int E) {
    int edge = blockIdx.x * 8 + (threadIdx.x >> 5);
    if (edge >= E) return;
    int lane = threadIdx.x & 31;
    int s = (int)src[edge];
    int d = (int)dst[edge];
    const float4 v = ((const float4*)(feat + (size_t)s * D))[lane];
    float* p = agg + (size_t)d * D + lane * 4;
    atomicAdd(p + 0, v.x);
    atomicAdd(p + 1, v.y);
    atomicAdd(p + 2, v.z);
    atomicAdd(p + 3, v.w);
    if (lane == 0) atomicAdd(&cnt[d], 1.0f);
}

// -------------------------------- fused SAGE GEMM + L2-norm (+ReLU+BN stats)
// block = 128 threads (4 waves), computes 16 nodes x 128 features.
// out[m][n] = (agg[m]/max(cnt,1)) . Wl[n] + bl[n] + x[m] . Wr[n], L2-normalized.
// A tiles staged LDS via GLOBAL_LOAD_ASYNC_TO_LDS_B128 (ASYNCcnt path);
// the 1/cnt row scaling is deferred to the epilogue (commutes with the GEMM).
__global__ __launch_bounds__(128)
void sage_gemm(const float* __restrict__ agg, const float* __restrict__ cnt,
               const float* __restrict__ xin,
               const float* __restrict__ Wl, const float* __restrict__ bl,
               const float* __restrict__ Wr,
               float* __restrict__ out,
               int do_bn, float* __restrict__ gsum, float* __restrict__ gsumsq) {
    __shared__ float Asha[16][132];   // raw neighbor-sum tile (padded stride)
    __shared__ float Ashx[16][132];   // root-feature tile
    __shared__ float rinv[16];
    __shared__ float rowSq[16];
    __shared__ float colS[D];
    __shared__ float colS2[D];

    const int tid  = threadIdx.x;
    const int lane = tid & 31;
    const int wave = tid >> 5;
    const size_t node0 = (size_t)blockIdx.x * 16;

    if (tid < 16) {
        rinv[tid]  = 1.0f / fmaxf(cnt[node0 + tid], 1.0f);
        rowSq[tid] = 0.0f;
    }
    if (do_bn) { colS[tid] = 0.0f; colS2[tid] = 0.0f; }

    // ---- async memory->LDS staging: 2 tiles x 16 rows x 128 floats ----
    // 512 x 16B chunks per tile; 128 threads -> 4 chunks each per tile.
    {
        const unsigned baseA = (unsigned)(uintptr_t)&Asha[0][0];
        const unsigned baseX = (unsigned)(uintptr_t)&Ashx[0][0];
#pragma unroll
        for (int j = 0; j < 4; ++j) {
            int i  = tid + j * 128;
            int r  = i >> 5;        // row 0..15
            int c4 = i & 31;        // 16B chunk within row
            unsigned loff = (unsigned)(r * (132 * 4) + c4 * 16);
            const float* ga = agg + (node0 + r) * D + c4 * 4;
            const float* gx = xin + (node0 + r) * D + c4 * 4;
            asm volatile("global_load_async_to_lds_b128 %0, %1, off"
                         :: "v"(baseA + loff), "v"(ga) : "memory");
            asm volatile("global_load_async_to_lds_b128 %0, %1, off"
                         :: "v"(baseX + loff), "v"(gx) : "memory");
        }
    }
    asm volatile("s_wait_asynccnt 0x0" ::: "memory");
    __syncthreads();

    const int m     = lane & 15;           // A-matrix row for this lane
    const int khalf = (lane >> 4) << 1;    // K sub-offset: 0 or 2
    const int rbase = (lane >> 4) << 3;    // C/D row base: 0 or 8

    v8f accL[2], accR[2];
#pragma unroll
    for (int t = 0; t < 2; ++t) {
        const int n = (wave * 2 + t) * 16 + (lane & 15);  // output feature
        float bv = bl[n];
        v8f c0, c1;
#pragma unroll
        for (int v = 0; v < 8; ++v) { c0[v] = 0.0f; c1[v] = bv; }
        const float* wlrow = Wl + (size_t)n * D;  // B[k][n] = W[n][k]
        const float* wrrow = Wr + (size_t)n * D;
#pragma unroll 8
        for (int k0 = 0; k0 < D; k0 += 4) {       // neighbor GEMM (unscaled)
            v2f a = *(const v2f*)&Asha[m][k0 + khalf];
            v2f b = *(const v2f*)&wlrow[k0 + khalf];
            c0 = __builtin_amdgcn_wmma_f32_16x16x4_f32(false, a, false, b,
                                                       (short)0, c0, false, false);
        }
#pragma unroll 8
        for (int k0 = 0; k0 < D; k0 += 4) {       // root GEMM (+bias)
            v2f a = *(const v2f*)&Ashx[m][k0 + khalf];
            v2f b = *(const v2f*)&wrrow[k0 + khalf];
            c1 = __builtin_amdgcn_wmma_f32_16x16x4_f32(false, a, false, b,
                                                       (short)0, c1, false, false);
        }
        accL[t] = c0;
        accR[t] = c1;
    }

    // combine (apply deferred mean scaling) + per-row sum of squares
#pragma unroll
    for (int t = 0; t < 2; ++t)
#pragma unroll
        for (int v = 0; v < 8; ++v) {
            int row = v + rbase;
            float val = accL[t][v] * rinv[row] + accR[t][v];
            accR[t][v] = val;                       // stash combined value
            atomicAdd(&rowSq[row], val * val);
        }
    __syncthreads();

    // L2-normalize, optional ReLU + BN-stat accumulation, write out
#pragma unroll
    for (int t = 0; t < 2; ++t) {
        const int n = (wave * 2 + t) * 16 + (lane & 15);
#pragma unroll
        for (int v = 0; v < 8; ++v) {
            int row  = v + rbase;
            float rs = 1.0f / fmaxf(sqrtf(rowSq[row]), L2_EPS);
            float val = accR[t][v] * rs;
            if (do_bn) {
                val = fmaxf(val, 0.0f);
                atomicAdd(&colS[n], val);
                atomicAdd(&colS2[n], val * val);
            }
            out[(node0 + row) * D + n] = val;
        }
    }
    if (do_bn) {
        __syncthreads();
        atomicAdd(&gsum[tid], colS[tid]);
        atomicAdd(&gsumsq[tid], colS2[tid]);
    }
}

// ---------------------------------------------------------------- BN
__global__ void bn_finalize(const float* __restrict__ gsum,
                            const float* __restrict__ gsumsq,
                            const float* __restrict__ gamma,
                            const float* __restrict__ beta,
                            float* __restrict__ scale, float* __restrict__ shift) {
    int c = threadIdx.x;
    const float invN = 1.0f / (float)N_NODES;
    float mu  = gsum[c] * invN;
    float var = gsumsq[c] * invN - mu * mu;
    float sc  = gamma[c] * rsqrtf(var + BN_EPS);
    scale[c] = sc;
    shift[c] = beta[c] - mu * sc;
}

__global__ void bn_apply(float* __restrict__ h, const float* __restrict__ scale,
                         const float* __restrict__ shift, size_t n) {
    size_t i = (size_t)blockIdx.x * blockDim.x + threadIdx.x;
    size_t stride = (size_t)gridDim.x * blockDim.x;
    for (; i < n; i += stride) {
        int c = (int)(i & (D - 1));
        h[i] = h[i] * scale[c] + shift[c];
    }
}

// ---------------------------------------------------------------- launch
extern "C" void kernel_launch(void* const* d_in, const int* in_sizes, int n_in,
                              void* d_out, int out_size, void* d_ws, size_t ws_size,
                              hipStream_t stream) {
    const float*     x     = (const float*)d_in[0];
    const long long* ei    = (const long long*)d_in[1];  // int64 [2,E]
    const float*     W1l   = (const float*)d_in[2];
    const float*     b1l   = (const float*)d_in[3];
    const float*     W1r   = (const float*)d_in[4];
    const float*     gamma = (const float*)d_in[5];
    const float*     beta  = (const float*)d_in[6];
    const float*     W2l   = (const float*)d_in[7];
    const float*     b2l   = (const float*)d_in[8];
    const float*     W2r   = (const float*)d_in[9];
    float* out = (float*)d_out;
    const int E = in_sizes[1] / 2;
    const long long* src = ei;
    const long long* dst = ei + E;

    // workspace layout (contiguous so one zero pass covers agg+cnt+stats)
    float* ws     = (float*)d_ws;
    float* agg    = ws;                              // N*D
    float* cnt    = agg + (size_t)N_NODES * D;       // N
    float* gsum   = cnt + N_NODES;                   // D
    float* gsumsq = gsum + D;                        // D
    float* scale  = gsumsq + D;                      // D
    float* shift  = scale + D;                       // D
    float* h1     = shift + D;                       // N*D

    const size_t ND = (size_t)N_NODES * D;

    // ---- layer 1 ----
    zero_kernel<<<1024, 256, 0, stream>>>(agg, ND + N_NODES + 2 * D);
    scatter_kernel<<<(E + 7) / 8, 256, 0, stream>>>(x, src, dst, agg, cnt, E);
    sage_gemm<<<N_NODES / 16, 128, 0, stream>>>(agg, cnt, x, W1l, b1l, W1r,
                                                h1, 1, gsum, gsumsq);
    // ---- transition: ReLU folded above; BN with batch stats ----
    bn_finalize<<<1, D, 0, stream>>>(gsum, gsumsq, gamma, beta, scale, shift);
    bn_apply<<<2048, 256, 0, stream>>>(h1, scale, shift, ND);

    // ---- layer 2 ----
    zero_kernel<<<1024, 256, 0, stream>>>(agg, ND + N_NODES);
    scatter_kernel<<<(E + 7) / 8, 256, 0, stream>>>(h1, src, dst, agg, cnt, E);
    sage_gemm<<<N_NODES / 16, 128, 0, stream>>>(agg, cnt, h1, W2l, b2l, W2r,
                                                out, 0, gsum, gsumsq);
}